// IDGCN_27479200760360
// MI455X (gfx1250) — compile-verified
//
#include <hip/hip_runtime.h>
#include <hip/hip_bf16.h>

typedef __attribute__((ext_vector_type(2))) float v2f;
typedef __attribute__((ext_vector_type(8))) float v8f;

// ---------------- degree / normalization ----------------

__global__ void deg_init_kernel(float* __restrict__ deg, int n) {
    int i = blockIdx.x * blockDim.x + threadIdx.x;
    if (i < n) deg[i] = 1.0f;  // self-loop weight (fill_weight = 1.0)
}

__global__ void deg_edges_kernel(const int* __restrict__ row,
                                 const float* __restrict__ w,
                                 float* __restrict__ deg, int E) {
    int e = blockIdx.x * blockDim.x + threadIdx.x;
    if (e < E) atomicAdd(&deg[row[e]], w[e]);
}

__global__ void dis_kernel(const float* __restrict__ deg,
                           float* __restrict__ dis, int n) {
    int i = blockIdx.x * blockDim.x + threadIdx.x;
    if (i < n) {
        float d = deg[i];
        dis[i] = (d > 0.0f) ? rsqrtf(d) : 0.0f;
    }
}

// ---------------- dense transform: h = x @ kernel (fp32 WMMA) ----------------
// One wave32 per 16x16 output tile. K=64 reduced via 16 chained
// V_WMMA_F32_16X16X4_F32 ops.
// A 16x4 layout (ISA 7.12.2): lane l (half = l>>4, m = l&15):
//   a[v] = A[m][k0 + 2*half + v]
// B 4x16 layout (mirror):    b[v] = B[k0 + 2*half + v][n]
// C/D 16x16: c[r] = D[r + 8*half][l&15]

__global__ void __launch_bounds__(256)
gemm_h_wmma(const float* __restrict__ x, const float* __restrict__ kern,
            float* __restrict__ h, int nodes) {
    int wave = blockIdx.x * (blockDim.x >> 5) + (threadIdx.x >> 5);
    int lane = threadIdx.x & 31;
    int tileM = wave >> 2;        // 4 N-tiles (64/16)
    int tileN = wave & 3;
    int tilesM = (nodes + 15) >> 4;
    if (tileM >= tilesM) return;  // wave-uniform: EXEC stays all-1s for WMMA

    int half = lane >> 4;
    int l16  = lane & 15;
    int rowA = tileM * 16 + l16;
    if (rowA >= nodes) rowA = nodes - 1;   // clamp loads; stores guarded below
    int colB = tileN * 16 + l16;

    const float* xr = x + (long long)rowA * 64;
    v8f c = {};
#pragma unroll
    for (int k0 = 0; k0 < 64; k0 += 4) {
        int ka = k0 + 2 * half;
        v2f a, b;
        a.x = xr[ka];
        a.y = xr[ka + 1];
        b.x = kern[(ka + 0) * 64 + colB];
        b.y = kern[(ka + 1) * 64 + colB];
        c = __builtin_amdgcn_wmma_f32_16x16x4_f32(false, a, false, b,
                                                  (short)0, c, false, false);
    }
    int mBase = tileM * 16 + 8 * half;
#pragma unroll
    for (int r = 0; r < 8; ++r) {
        int m = mBase + r;
        if (m < nodes) h[(long long)m * 64 + tileN * 16 + l16] = c[r];
    }
}

// ---------------- id transform: h[id] += x[id] @ kernel_id (WMMA + atomics) --

__global__ void __launch_bounds__(256)
gemm_id_wmma(const float* __restrict__ x, const int* __restrict__ id_index,
             const float* __restrict__ kern_id, float* __restrict__ h,
             int nid) {
    int wave = blockIdx.x * (blockDim.x >> 5) + (threadIdx.x >> 5);
    int lane = threadIdx.x & 31;
    int tileI = wave >> 2;
    int tileN = wave & 3;
    int tilesI = (nid + 15) >> 4;
    if (tileI >= tilesI) return;  // wave-uniform

    int half = lane >> 4;
    int l16  = lane & 15;
    int p = tileI * 16 + l16;
    if (p >= nid) p = nid - 1;
    int node = id_index[p];
    int colB = tileN * 16 + l16;

    const float* xr = x + (long long)node * 64;
    v8f c = {};
#pragma unroll
    for (int k0 = 0; k0 < 64; k0 += 4) {
        int ka = k0 + 2 * half;
        v2f a, b;
        a.x = xr[ka];
        a.y = xr[ka + 1];
        b.x = kern_id[(ka + 0) * 64 + colB];
        b.y = kern_id[(ka + 1) * 64 + colB];
        c = __builtin_amdgcn_wmma_f32_16x16x4_f32(false, a, false, b,
                                                  (short)0, c, false, false);
    }
#pragma unroll
    for (int r = 0; r < 8; ++r) {
        int pr = tileI * 16 + r + 8 * half;
        if (pr < nid) {
            int n2 = id_index[pr];  // id_index may repeat -> atomic scatter-add
            atomicAdd(&h[(long long)n2 * 64 + tileN * 16 + l16], c[r]);
        }
    }
}

// ---------------- out init: self-loop term + bias ----------------

__global__ void out_init_kernel(const float* __restrict__ dis,
                                const float* __restrict__ h,
                                const float* __restrict__ bias,
                                float* __restrict__ out, int n) {
    long long tid = (long long)blockIdx.x * blockDim.x + threadIdx.x;
    long long i = tid >> 6;
    int d = (int)(tid & 63);
    if (i < n) {
        float s = dis[i];
        out[i * 64 + d] = s * s * h[i * 64 + d] + bias[d];
    }
}

// ---------------- edge SpMM: out[row] += w_norm * h[col] ----------------
// 64 threads per edge: coalesced 64-wide gather of h[col], atomic scatter
// to out[row]. h (25.6 MB) resides in the 192 MB L2.

__global__ void spmm_edges_kernel(const int* __restrict__ row,
                                  const int* __restrict__ col,
                                  const float* __restrict__ w,
                                  const float* __restrict__ dis,
                                  const float* __restrict__ h,
                                  float* __restrict__ out, int E) {
    long long tid = (long long)blockIdx.x * blockDim.x + threadIdx.x;
    long long e = tid >> 6;
    int d = (int)(tid & 63);
    if (e >= E) return;
    int r  = row[e];
    int cn = col[e];
    float wn = dis[r] * w[e] * dis[cn];
    atomicAdd(&out[(long long)r * 64 + d], wn * h[(long long)cn * 64 + d]);
}

// ---------------- launch ----------------

extern "C" void kernel_launch(void* const* d_in, const int* in_sizes, int n_in,
                              void* d_out, int out_size, void* d_ws, size_t ws_size,
                              hipStream_t stream) {
    const float* x        = (const float*)d_in[0];
    const int*   eidx     = (const int*)d_in[1];
    const int*   id_index = (const int*)d_in[2];
    const float* ew       = (const float*)d_in[3];
    const float* kern     = (const float*)d_in[4];
    const float* kern_id  = (const float*)d_in[5];
    const float* bias     = (const float*)d_in[6];
    float* out = (float*)d_out;

    const int N   = in_sizes[0] / 64;   // 100000
    const int E   = in_sizes[3];        // 1600000
    const int NID = in_sizes[2];        // 10000
    const int* row = eidx;              // edge_index[0]
    const int* col = eidx + E;          // edge_index[1]

    // workspace: h [N*64] | deg [N] | dis [N]
    float* h   = (float*)d_ws;
    float* deg = h + (size_t)N * 64;
    float* dis = deg + N;

    // 1) degrees (self-loop=1.0 baked into init), 2) rsqrt
    deg_init_kernel<<<(N + 255) / 256, 256, 0, stream>>>(deg, N);
    deg_edges_kernel<<<(E + 255) / 256, 256, 0, stream>>>(row, ew, deg, E);
    dis_kernel<<<(N + 255) / 256, 256, 0, stream>>>(deg, dis, N);

    // 3) h = x @ kernel  (fp32 WMMA, one wave per 16x16 tile)
    int tilesM = (N + 15) / 16;
    int wavesH = tilesM * 4;
    gemm_h_wmma<<<(wavesH + 7) / 8, 256, 0, stream>>>(x, kern, h, N);

    // 4) h[id] += x[id] @ kernel_id
    int tilesI = (NID + 15) / 16;
    int wavesI = tilesI * 4;
    gemm_id_wmma<<<(wavesI + 7) / 8, 256, 0, stream>>>(x, id_index, kern_id, h, NID);

    // 5) out = dis^2 * h + bias (self-loop contribution)
    long long tOut = (long long)N * 64;
    out_init_kernel<<<(unsigned)((tOut + 255) / 256), 256, 0, stream>>>(dis, h, bias, out, N);

    // 6) edge scatter: out[row] += dis[row]*w*dis[col] * h[col]
    long long tE = (long long)E * 64;
    spmm_edges_kernel<<<(unsigned)((tE + 255) / 256), 256, 0, stream>>>(row, col, ew, dis, h, out, E);
}